// GCN_29094108463582
// MI455X (gfx1250) — compile-verified
//
#include <hip/hip_runtime.h>
#include <cstdint>

// ---------------------------------------------------------------------------
// GCN forward for MI455X (gfx1250):
//   - fp32 WMMA GEMMs with double-buffered LDS tiles, A staged via
//     global_load_async_to_lds_b128 (ASYNCcnt), B staged as packed float2
//   - L2-resident edge gather/scatter with global_atomic_add_f32
// ---------------------------------------------------------------------------

typedef __attribute__((ext_vector_type(2))) float v2f;
typedef __attribute__((ext_vector_type(8))) float v8f;

static constexpr int   HD    = 128;    // hidden width
static constexpr float BNEPS = 1e-5f;
static constexpr int   KT    = 32;     // K tile for GEMM staging
static constexpr int   ASTR  = 36;     // padded LDS row stride for A (floats)

// ---- CDNA5 async global->LDS copy (16B per lane), inline asm (portable) ----
static __device__ __forceinline__ void async_g2l_b128(const float* g, void* l) {
    unsigned lds_off = (unsigned)(uintptr_t)l;   // flat shared addr: low 32 bits = LDS offset
    asm volatile("global_load_async_to_lds_b128 %0, %1, off"
                 :: "v"(lds_off), "v"(g) : "memory");
}
static __device__ __forceinline__ void wait_async0() {
    asm volatile("s_wait_asynccnt 0x0" ::: "memory");
}

// ---------------- degree / norm ----------------

__global__ void k_init_deg(float* __restrict__ deg, int n) {
    int i = blockIdx.x * blockDim.x + threadIdx.x;
    if (i < n) deg[i] = 1.0f;   // self-loop weight folded in
}

__global__ void k_deg(const long long* __restrict__ col,
                      const float* __restrict__ ew,
                      float* __restrict__ deg, long long E) {
    long long e = (long long)blockIdx.x * blockDim.x + threadIdx.x;
    if (e < E) atomicAdd(&deg[col[e]], ew[e]);
}

__global__ void k_norm(const long long* __restrict__ row,
                       const long long* __restrict__ col,
                       const float* __restrict__ ew,
                       const float* __restrict__ deg,
                       float* __restrict__ nrm, long long E) {
    long long e = (long long)blockIdx.x * blockDim.x + threadIdx.x;
    if (e < E) {
        float dr = deg[row[e]];
        float dc = deg[col[e]];
        nrm[e] = rsqrtf(dr) * ew[e] * rsqrtf(dc);   // deg >= 1 always (self loop)
    }
}

__global__ void k_selfnorm(float* __restrict__ deg, int n) {
    int i = blockIdx.x * blockDim.x + threadIdx.x;
    if (i < n) deg[i] = 1.0f / deg[i];   // dis[i]^2 for the self-loop edge
}

// ---------------- fp32 WMMA GEMM: C[N x 128] = A[N x K] @ W[K x 128] -------
// Block = 4 waves = 64 rows x 128 cols. K tiled by 32, double-buffered LDS.
// A tile (64x32) arrives via async global->LDS b128; B tile stored as
// packed float2 {W[k][n], W[k+1][n]} so each B fragment is one ds_load_b64.
// Frag layouts (f32 16x16x4): lanes 0-15 hold K=0,1; lanes 16-31 hold K=2,3.

__global__ void __launch_bounds__(128)
k_gemm_wmma(const float* __restrict__ A, const float* __restrict__ W,
            float* __restrict__ C, int N, int K) {
    __shared__ __align__(16) float  Asmem[2][64][ASTR];
    __shared__ __align__(16) float2 Bsmem[2][16][HD + 2];

    const int tid  = threadIdx.x;
    const int wv   = tid >> 5;
    const int lane = tid & 31;
    const int m    = lane & 15;
    const int hi   = lane >> 4;
    const int row0 = blockIdx.x * 64;
    const long long Nm1 = N - 1;

    const int nT = K / KT;

    // ---- stagers ----
    auto stageA = [&](int nb, int kt) {
#pragma unroll
        for (int j = 0; j < 4; ++j) {
            int idx = tid + j * 128;
            int r   = idx >> 3;
            int c4  = idx & 7;
            long long gr = (long long)row0 + r;
            if (gr > Nm1) gr = Nm1;                  // clamp OOB tail rows
            const float* g = A + (size_t)gr * K + kt + c4 * 4;
            async_g2l_b128(g, &Asmem[nb][r][c4 * 4]);
        }
    };
    auto stageB = [&](int nb, int kt) {
#pragma unroll
        for (int j = 0; j < 16; ++j) {
            int idx = tid + j * 128;
            int p   = idx >> 7;
            int n   = idx & 127;
            float2 v;
            v.x = W[(size_t)(kt + 2 * p) * HD + n];
            v.y = W[(size_t)(kt + 2 * p + 1) * HD + n];
            Bsmem[nb][p][n] = v;
        }
    };

    const v8f vz = {0.f, 0.f, 0.f, 0.f, 0.f, 0.f, 0.f, 0.f};
    v8f acc[8];
#pragma unroll
    for (int t = 0; t < 8; ++t) acc[t] = vz;

    stageA(0, 0);
    stageB(0, 0);
    wait_async0();
    __syncthreads();

    const int rloc = wv * 16 + m;   // local A row this lane contributes

    for (int tile = 0;; ++tile) {
        const int cur = tile & 1;
        if (tile + 1 < nT) {
            stageA(cur ^ 1, (tile + 1) * KT);
            stageB(cur ^ 1, (tile + 1) * KT);
        }
#pragma unroll
        for (int s = 0; s < KT / 4; ++s) {
            float2 af = *(const float2*)&Asmem[cur][rloc][s * 4 + 2 * hi];
            v2f a; a.x = af.x; a.y = af.y;
#pragma unroll
            for (int t = 0; t < 8; ++t) {
                float2 bf = Bsmem[cur][2 * s + hi][t * 16 + m];
                v2f b; b.x = bf.x; b.y = bf.y;
                acc[t] = __builtin_amdgcn_wmma_f32_16x16x4_f32(
                    false, a, false, b, (short)0, acc[t], false, false);
            }
        }
        if (tile + 1 >= nT) break;
        wait_async0();
        __syncthreads();
    }

    // wave-uniform store guard (keeps EXEC all-ones through the WMMA loop)
    if (row0 + wv * 16 + 16 <= N) {
        float* Cp = C + (size_t)(row0 + wv * 16) * HD;
#pragma unroll
        for (int t = 0; t < 8; ++t) {
#pragma unroll
            for (int r = 0; r < 8; ++r) {
                Cp[(size_t)(r + 8 * hi) * HD + t * 16 + m] = acc[t][r];
            }
        }
    }
}

// ---------------- aggregation (message passing) ----------------

// t[i,:] = selfnorm[i]*h[i,:] + bias[:] (+ resid[i,:])   -- bias is post-agg
__global__ void k_agg_init(const float* __restrict__ h,
                           const float* __restrict__ selfn,
                           const float* __restrict__ bias,
                           const float* __restrict__ resid,
                           float* __restrict__ t, long long NH) {
    long long i = (long long)blockIdx.x * blockDim.x + threadIdx.x;
    if (i >= NH) return;
    int node = (int)(i >> 7);
    int j    = (int)(i & (HD - 1));
    float v = selfn[node] * h[i] + bias[j];
    if (resid) v += resid[i];
    t[i] = v;
}

// t[col,:] += norm[e] * h[row,:]
// 32 lanes per edge, float4 per lane: coalesced 512B gather + 4 f32 atomics.
__global__ void __launch_bounds__(256)
k_agg_edges(const long long* __restrict__ row,
            const long long* __restrict__ col,
            const float* __restrict__ nrm,
            const float* __restrict__ h,
            float* __restrict__ t, long long E) {
    long long e = (long long)blockIdx.x * 8 + (threadIdx.x >> 5);
    if (e >= E) return;
    int lane = threadIdx.x & 31;
    long long r = row[e];
    long long c = col[e];
    float w = nrm[e];
    const float4* hp = (const float4*)(h + r * HD);
    float4 hv = hp[lane];
    float* tp = t + c * HD + lane * 4;
    atomicAdd(tp + 0, w * hv.x);
    atomicAdd(tp + 1, w * hv.y);
    atomicAdd(tp + 2, w * hv.z);
    atomicAdd(tp + 3, w * hv.w);
}

// ---------------- batchnorm(eval) + relu, in place ----------------

__global__ void k_bn_relu(float* __restrict__ t,
                          const float* __restrict__ g,
                          const float* __restrict__ be,
                          const float* __restrict__ mean,
                          const float* __restrict__ var, long long NH) {
    long long i = (long long)blockIdx.x * blockDim.x + threadIdx.x;
    if (i >= NH) return;
    int j = (int)(i & (HD - 1));
    float v = (t[i] - mean[j]) * rsqrtf(var[j] + BNEPS) * g[j] + be[j];
    t[i] = v > 0.f ? v : 0.f;
}

// ---------------- final projection: z[i] = dot(t[i,:], W3) ----------------

__global__ void __launch_bounds__(32)
k_proj(const float* __restrict__ t, const float* __restrict__ W3,
       float* __restrict__ z, int N) {
    int i    = blockIdx.x;
    int lane = threadIdx.x;
    if (i >= N) return;
    const float4* tp = (const float4*)(t + (size_t)i * HD);
    const float4* wp = (const float4*)W3;
    float4 a = tp[lane];
    float4 b = wp[lane];
    float s = a.x * b.x + a.y * b.y + a.z * b.z + a.w * b.w;
#pragma unroll
    for (int off = 16; off > 0; off >>= 1) s += __shfl_xor(s, off, 32);
    if (lane == 0) z[i] = s;
}

__global__ void k_out_init(const float* __restrict__ z,
                           const float* __restrict__ selfn,
                           const float* __restrict__ b3,
                           float* __restrict__ out, int N) {
    int i = blockIdx.x * blockDim.x + threadIdx.x;
    if (i < N) out[i] = selfn[i] * z[i] + b3[0];
}

__global__ void k_out_edges(const long long* __restrict__ row,
                            const long long* __restrict__ col,
                            const float* __restrict__ nrm,
                            const float* __restrict__ z,
                            float* __restrict__ out, long long E) {
    long long e = (long long)blockIdx.x * blockDim.x + threadIdx.x;
    if (e < E) atomicAdd(&out[col[e]], nrm[e] * z[row[e]]);
}

// ---------------------------------------------------------------------------

extern "C" void kernel_launch(void* const* d_in, const int* in_sizes, int n_in,
                              void* d_out, int out_size, void* d_ws, size_t ws_size,
                              hipStream_t stream) {
    const float*     x    = (const float*)d_in[0];
    const long long* ei   = (const long long*)d_in[1];
    const float*     ew   = (const float*)d_in[2];
    const float*     W1   = (const float*)d_in[3];
    const float*     b1   = (const float*)d_in[4];
    const float*     W2   = (const float*)d_in[5];
    const float*     b2   = (const float*)d_in[6];
    const float*     W3   = (const float*)d_in[7];
    const float*     b3   = (const float*)d_in[8];
    const float*     Wres = (const float*)d_in[9];
    const float*     g1   = (const float*)d_in[10];
    const float*     be1  = (const float*)d_in[11];
    const float*     m1   = (const float*)d_in[12];
    const float*     v1   = (const float*)d_in[13];
    const float*     g2   = (const float*)d_in[14];
    const float*     be2  = (const float*)d_in[15];
    const float*     m2   = (const float*)d_in[16];
    const float*     v2   = (const float*)d_in[17];

    const int       N  = out_size;                 // 100000
    const long long E  = in_sizes[2];              // 1600000
    const int       Hh = in_sizes[4];              // 128
    const int       V  = in_sizes[3] / Hh;         // 256
    const long long NH = (long long)N * Hh;

    const long long* row  = ei;
    const long long* colp = ei + E;

    // workspace layout (floats)
    float* ws   = (float*)d_ws;
    float* deg  = ws;                         // N   (later: selfnorm)
    float* nrm  = deg + N;                    // E
    float* hbuf = nrm + E;                    // N*H (pre-aggregation GEMM out)
    float* tbuf = hbuf + NH;                  // N*H (aggregated)
    float* resb = tbuf + NH;                  // N*H (residual branch)
    float* zbuf = resb + NH;                  // N   (final projection)

    float* out = (float*)d_out;

    const int gemmBlocks = (N + 63) / 64;

    dim3 b256(256);
    int  gN  = (N + 255) / 256;
    int  gE  = (int)((E + 255) / 256);
    int  gNH = (int)((NH + 255) / 256);
    int  gE8 = (int)((E + 7) / 8);

    // --- normalization coefficients ---
    k_init_deg<<<gN, b256, 0, stream>>>(deg, N);
    k_deg<<<gE, b256, 0, stream>>>(colp, ew, deg, E);
    k_norm<<<gE, b256, 0, stream>>>(row, colp, ew, deg, nrm, E);
    k_selfnorm<<<gN, b256, 0, stream>>>(deg, N);      // deg := 1/deg (self-loop norm)

    // --- residual branch: res = x @ Wres ---
    k_gemm_wmma<<<gemmBlocks, 128, 0, stream>>>(x, Wres, resb, N, V);

    // --- layer 1: h = x @ W1 ; aggregate ; bn+relu ---
    k_gemm_wmma<<<gemmBlocks, 128, 0, stream>>>(x, W1, hbuf, N, V);
    k_agg_init<<<gNH, b256, 0, stream>>>(hbuf, deg, b1, nullptr, tbuf, NH);
    k_agg_edges<<<gE8, b256, 0, stream>>>(row, colp, nrm, hbuf, tbuf, E);
    k_bn_relu<<<gNH, b256, 0, stream>>>(tbuf, g1, be1, m1, v1, NH);

    // --- layer 2: h = t @ W2 ; aggregate (+res) ; bn+relu ---
    k_gemm_wmma<<<gemmBlocks, 128, 0, stream>>>(tbuf, W2, hbuf, N, Hh);
    k_agg_init<<<gNH, b256, 0, stream>>>(hbuf, deg, b2, resb, tbuf, NH);
    k_agg_edges<<<gE8, b256, 0, stream>>>(row, colp, nrm, hbuf, tbuf, E);
    k_bn_relu<<<gNH, b256, 0, stream>>>(tbuf, g2, be2, m2, v2, NH);

    // --- layer 3: z = t @ W3 ; aggregate into out ---
    k_proj<<<N, 32, 0, stream>>>(tbuf, W3, zbuf, N);
    k_out_init<<<gN, b256, 0, stream>>>(zbuf, deg, b3, out, N);
    k_out_edges<<<gE, b256, 0, stream>>>(row, colp, nrm, zbuf, out, E);
}